// GNN_90769838833826
// MI455X (gfx1250) — compile-verified
//
#include <hip/hip_runtime.h>

// ---------------------------------------------------------------------------
// GNN (GCN-style) forward for MI455X / gfx1250, wave32, WMMA f32 16x16x4.
//
//   deg -> dis -> norm (once)
//   h = x @ node_W + node_b                         (WMMA GEMM, K=64)
//   for l in 0..4:
//     hl = [relu?](h) @ lin_W[l] + lin_b[l]         (WMMA GEMM, K=128)
//     memset(dst = h or d_out)
//     message: tile WMMA  ee = ea @ edge_W[l],
//              val = (ee + edge_b + hl[row]) * norm, atomicAdd -> dst
//
// A-tiles staged in LDS (float4, once per block, shared by 8 waves);
// K-loops fully unrolled, branch-free; gather/scatter use uint32 element
// offsets so the backend emits SGPR-base + VGPR-offset (GVS) addressing.
// ---------------------------------------------------------------------------

typedef __attribute__((ext_vector_type(2))) float v2f;
typedef __attribute__((ext_vector_type(8))) float v8f;

static __device__ __forceinline__ v8f wmma_f32_16x16x4(v2f a, v2f b, v8f c) {
  // (neg_a, A, neg_b, B, c_mod, C, reuse_a, reuse_b)
  return __builtin_amdgcn_wmma_f32_16x16x4_f32(false, a, false, b, (short)0, c,
                                               false, false);
}

#define EMB 128

// ---------------- degree / normalization ----------------

__global__ __launch_bounds__(256) void k_init_deg(float* __restrict__ deg, int n) {
  int i = blockIdx.x * 256 + threadIdx.x;
  if (i < n) deg[i] = 1.0f;  // self loop contributes 1
}

__global__ __launch_bounds__(256) void k_accum_deg(const long long* __restrict__ rows,
                                                   float* __restrict__ deg, int E) {
  int e = blockIdx.x * 256 + threadIdx.x;
  if (e < E) atomicAdd(&deg[(unsigned)rows[e]], 1.0f);
}

__global__ __launch_bounds__(256) void k_dis(float* __restrict__ deg, int n) {
  int i = blockIdx.x * 256 + threadIdx.x;
  if (i < n) {
    float d = deg[i];
    deg[i] = (d > 0.0f) ? rsqrtf(fmaxf(d, 1e-30f)) : 0.0f;
  }
}

__global__ __launch_bounds__(256) void k_norm(const long long* __restrict__ ei,
                                              const float* __restrict__ dis,
                                              float* __restrict__ nrm, int E, int ET) {
  int e = blockIdx.x * 256 + threadIdx.x;
  if (e >= ET) return;
  unsigned r, c;
  if (e < E) { r = (unsigned)ei[e]; c = (unsigned)ei[(size_t)E + e]; }
  else       { r = c = (unsigned)(e - E); }  // self loop
  nrm[e] = dis[r] * dis[c];
}

// ---------------- dense GEMM: out[M,128] = (relu?)A[M,K] @ W[K,128] + bias ---
// Block = 256 threads = 8 waves; block computes a 16 x 128 output tile.

template <int K, bool RELU>
__global__ __launch_bounds__(256) void k_gemm_wmma(const float* __restrict__ A,
                                                   const float* __restrict__ W,
                                                   const float* __restrict__ bias,
                                                   float* __restrict__ out,
                                                   int M) {
  constexpr int KP = K + 4;  // stride 132/68 words: 16B-aligned b128 stores,
  __shared__ float As[16 * KP];  // 8B-aligned b64 reads, banks spread by 4*m

  const int tid  = threadIdx.x;
  const int lane = tid & 31;
  const int wave = tid >> 5;
  const int half = (lane >= 16);
  const int ml   = lane & 15;
  const int m0   = blockIdx.x * 16;
  const int gcol = wave * 16 + ml;
  const bool fullTile = (m0 + 16 <= M);

  // ---- stage A tile (16 x K) into LDS as float4; relu folded in ----
  constexpr int NV = 16 * K / 4;
  if (fullTile) {
#pragma unroll
    for (int idx = tid; idx < NV; idx += 256) {
      const int r = (idx * 4) / K, c = (idx * 4) % K;
      float4 v = *(const float4*)&A[(size_t)(m0 + r) * K + c];
      if (RELU) {
        v.x = fmaxf(v.x, 0.0f); v.y = fmaxf(v.y, 0.0f);
        v.z = fmaxf(v.z, 0.0f); v.w = fmaxf(v.w, 0.0f);
      }
      *(float4*)&As[r * KP + c] = v;
    }
  } else {
#pragma unroll
    for (int idx = tid; idx < NV; idx += 256) {
      const int r = (idx * 4) / K, c = (idx * 4) % K;
      float4 v = make_float4(0.f, 0.f, 0.f, 0.f);
      if (m0 + r < M) {
        v = *(const float4*)&A[(size_t)(m0 + r) * K + c];
        if (RELU) {
          v.x = fmaxf(v.x, 0.0f); v.y = fmaxf(v.y, 0.0f);
          v.z = fmaxf(v.z, 0.0f); v.w = fmaxf(v.w, 0.0f);
        }
      }
      *(float4*)&As[r * KP + c] = v;
    }
  }
  __syncthreads();

  // ---- branch-free fully-unrolled WMMA K-loop ----
  v8f acc = {};
#pragma unroll
  for (int k0 = 0; k0 < K; k0 += 4) {
    const int ka = k0 + (half ? 2 : 0);
    v2f a = *(const v2f*)&As[ml * KP + ka];  // ds_load_b64
    v2f b;
    b.x = W[(unsigned)(ka * EMB + gcol)];        // saddr + voffset + imm
    b.y = W[(unsigned)((ka + 1) * EMB + gcol)];
    acc = wmma_f32_16x16x4(a, b, acc);
  }

  const float bv = bias[gcol];
  if (fullTile) {
#pragma unroll
    for (int r = 0; r < 8; ++r)
      out[(size_t)(m0 + r + (half ? 8 : 0)) * EMB + gcol] = acc[r] + bv;
  } else {
#pragma unroll
    for (int r = 0; r < 8; ++r) {
      const int row = m0 + r + (half ? 8 : 0);
      if (row < M) out[(size_t)row * EMB + gcol] = acc[r] + bv;
    }
  }
}

// ---------------- fused edge-MLP + gather + scale + scatter-add --------------
// Block = 256 threads = 8 waves; block handles a tile of 16 edges x 128 cols.

__global__ __launch_bounds__(256) void k_message_wmma(const long long* __restrict__ ei,
                                                      const float* __restrict__ ea,
                                                      const float* __restrict__ eW,
                                                      const float* __restrict__ eB,
                                                      const float* __restrict__ hl,
                                                      const float* __restrict__ nrm,
                                                      float* __restrict__ hout,
                                                      int E, int ET) {
  constexpr int KP = 18;             // 16 + 2 pad (b64 reads stay 8B aligned)
  __shared__ float Eas[16 * KP];     // edge-attr tile (self loops synthesized)
  __shared__ uint4 Smeta[16];        // {row*EMB, col*EMB, norm-bits, -}

  const int tid  = threadIdx.x;
  const int lane = tid & 31;
  const int wave = tid >> 5;
  const int half = (lane >= 16);
  const int ml   = lane & 15;
  const int e0   = blockIdx.x * 16;
  const int gcol = wave * 16 + ml;

  // ---- stage edge-attr tile + packed per-edge metadata ----
  {
    const int r = tid >> 4, k = tid & 15;
    const int e = e0 + r;
    float v = 0.0f;
    if (e < E)       v = ea[(size_t)e * 16 + k];
    else if (e < ET) v = (k == 0) ? 1.0f : 0.0f;  // self-loop one-hot attr
    Eas[r * KP + k] = v;
  }
  if (tid < 16) {
    const int e = e0 + tid;
    unsigned r = 0, c = 0;
    float nv = 0.0f;
    if (e < ET) {
      if (e < E) { r = (unsigned)ei[e]; c = (unsigned)ei[(size_t)E + e]; }
      else       { r = c = (unsigned)(e - E); }
      nv = nrm[e];
    }
    // store element offsets (row*128 fits easily in u32)
    Smeta[tid] = make_uint4(r * EMB, c * EMB, __float_as_uint(nv), 0u);
  }
  __syncthreads();

  // ---- ee-tile[16 x 16] = ea-tile[16 x 16] @ edge_W[16 x cols] ----
  v8f acc = {};
#pragma unroll
  for (int k0 = 0; k0 < 16; k0 += 4) {
    const int ka = k0 + (half ? 2 : 0);
    v2f a = *(const v2f*)&Eas[ml * KP + ka];  // ds_load_b64
    v2f b;
    b.x = eW[(unsigned)(ka * EMB + gcol)];
    b.y = eW[(unsigned)((ka + 1) * EMB + gcol)];
    acc = wmma_f32_16x16x4(a, b, acc);
  }

  // ---- msg = norm * (ee + bias + hl[row]) ; scatter-add to hout[col] ----
  const float bv = eB[gcol];
  unsigned ro[8], co[8];
  float nv[8];
#pragma unroll
  for (int r = 0; r < 8; ++r) {
    const uint4 md = Smeta[r + (half ? 8 : 0)];     // LDS broadcast
    ro[r] = md.x + (unsigned)gcol;                  // u32 element offsets ->
    co[r] = md.y + (unsigned)gcol;                  // GVS addressing, no 64-bit math
    nv[r] = __uint_as_float(md.z);
  }

  if (e0 + 16 <= ET) {                 // block-uniform full-tile fast path
    float g[8];
#pragma unroll
    for (int r = 0; r < 8; ++r)        // batched, clauseable gathers
      g[r] = hl[ro[r]];
#pragma unroll
    for (int r = 0; r < 8; ++r)
      atomicAdd(&hout[co[r]], (acc[r] + bv + g[r]) * nv[r]);
  } else {
#pragma unroll
    for (int r = 0; r < 8; ++r) {
      if (e0 + r + (half ? 8 : 0) < ET) {
        atomicAdd(&hout[co[r]], (acc[r] + bv + hl[ro[r]]) * nv[r]);
      }
    }
  }
}

// ---------------------------------------------------------------------------

extern "C" void kernel_launch(void* const* d_in, const int* in_sizes, int n_in,
                              void* d_out, int out_size, void* d_ws, size_t ws_size,
                              hipStream_t stream) {
  const float*     x         = (const float*)d_in[0];
  const long long* ei        = (const long long*)d_in[1];  // int64 [2,E]
  const float*     edge_attr = (const float*)d_in[2];
  const float*     node_W    = (const float*)d_in[3];
  const float*     node_b    = (const float*)d_in[4];
  const float*     lin_W     = (const float*)d_in[5];
  const float*     lin_b     = (const float*)d_in[6];
  const float*     edge_W    = (const float*)d_in[7];
  const float*     edge_b    = (const float*)d_in[8];
  (void)n_in; (void)out_size; (void)ws_size;

  const int NF  = 64;
  const int EAD = 16;
  const int L   = 5;
  const int N   = in_sizes[0] / NF;   // 50000
  const int E   = in_sizes[1] / 2;    // 640000
  const int ET  = E + N;

  // workspace carve-up (256B aligned): dis | norm | h | hl   (~54 MB)
  char*  ws  = (char*)d_ws;
  size_t off = 0;
  auto carve = [&](size_t bytes) -> void* {
    void* p = ws + off;
    off += (bytes + 255) & ~(size_t)255;
    return p;
  };
  float* dis = (float*)carve((size_t)N * sizeof(float));
  float* nrm = (float*)carve((size_t)ET * sizeof(float));
  float* h   = (float*)carve((size_t)N * EMB * sizeof(float));
  float* hl  = (float*)carve((size_t)N * EMB * sizeof(float));

  const dim3 blk(256);

  // --- symmetric degree normalization (once; reused by all layers) ---
  k_init_deg <<<(N  + 255) / 256, blk, 0, stream>>>(dis, N);
  k_accum_deg<<<(E  + 255) / 256, blk, 0, stream>>>(ei, dis, E);
  k_dis      <<<(N  + 255) / 256, blk, 0, stream>>>(dis, N);
  k_norm     <<<(ET + 255) / 256, blk, 0, stream>>>(ei, dis, nrm, E, ET);

  // --- node encoder: h = x @ node_W + node_b ---
  k_gemm_wmma<64, false><<<(N + 15) / 16, blk, 0, stream>>>(x, node_W, node_b, h, N);

  // --- message-passing layers ---
  for (int l = 0; l < L; ++l) {
    const float* lW = lin_W + (size_t)l * EMB * EMB;
    const float* lb = lin_b + (size_t)l * EMB;
    if (l > 0)
      k_gemm_wmma<128, true ><<<(N + 15) / 16, blk, 0, stream>>>(h, lW, lb, hl, N);
    else
      k_gemm_wmma<128, false><<<(N + 15) / 16, blk, 0, stream>>>(h, lW, lb, hl, N);

    // h is dead once hl exists; reuse as scatter target (d_out on last layer)
    float* dst = (l == L - 1) ? (float*)d_out : h;
    hipMemsetAsync(dst, 0, (size_t)N * EMB * sizeof(float), stream);

    k_message_wmma<<<(ET + 15) / 16, blk, 0, stream>>>(
        ei, edge_attr, edge_W + (size_t)l * EAD * EMB, edge_b + (size_t)l * EMB,
        hl, nrm, dst, E, ET);
  }
}